// DotProductPredictor_33328946217289
// MI455X (gfx1250) — compile-verified
//
#include <hip/hip_runtime.h>

typedef __attribute__((ext_vector_type(2))) float v2f;
typedef __attribute__((ext_vector_type(8))) float v8f;

#define D_FEAT 64
#define WAVES_PER_BLOCK 8
#define EDGES_PER_WAVE 16

// One wave32 computes 16 edge dot-products as the diagonal of a 16x16 f32
// WMMA accumulation over K=64 (16 steps of V_WMMA_F32_16X16X4_F32).
__global__ __launch_bounds__(256) void edge_dot_wmma_f32(
    const float* __restrict__ h,
    const int* __restrict__ src,
    const int* __restrict__ dst,
    float* __restrict__ out,
    int n_edges)
{
    const int lane = threadIdx.x & 31;            // wave32 lane id
    const int wave = threadIdx.x >> 5;            // wave index in block
    const long long tile = (long long)blockIdx.x * WAVES_PER_BLOCK + wave;
    const long long e_base = tile * EDGES_PER_WAVE;

    // Wave-uniform guard: all lanes of a wave agree, EXEC is all-1s inside.
    if (e_base < (long long)n_edges) {
        const int m    = lane & 15;               // edge slot / matrix row-col
        const int koff = (lane < 16) ? 0 : 2;     // K sub-offset per half-wave

        long long e = e_base + m;
        // Clamp tail edges to edge 0; their results are masked at store time.
        long long ec = (e < (long long)n_edges) ? e : 0;

        const int si = src[ec];
        const int di = dst[ec];

        // A: lane holds row M=si, elements (4t+koff, 4t+koff+1)
        // B: lane holds col N=di, same K split (layouts mirror per ISA 7.12.2)
        const float* aptr = h + (long long)si * D_FEAT + koff;
        const float* bptr = h + (long long)di * D_FEAT + koff;

        v8f acc = {};
#pragma unroll
        for (int t = 0; t < 16; ++t) {
            v2f a = *(const v2f*)(aptr + t * 4);   // global_load_b64, imm offset
            v2f b = *(const v2f*)(bptr + t * 4);
            // 8 args: (neg_a, A, neg_b, B, c_mod, C, reuse_a, reuse_b)
            acc = __builtin_amdgcn_wmma_f32_16x16x4_f32(
                false, a, false, b, (short)0, acc, false, false);
        }

        // Diagonal extraction D[m][m]:
        //   lanes 0..7   -> edge m = lane,      VGPR r = lane
        //   lanes 24..31 -> edge m = lane - 16, VGPR r = lane - 24
        const int r = (lane < 16) ? lane : (lane - 24);
        float diag = 0.0f;
#pragma unroll
        for (int v = 0; v < 8; ++v) diag = (r == v) ? acc[v] : diag;

        const bool writer = (lane < 8) || (lane >= 24);
        const long long eout = e_base + ((lane < 16) ? lane : (lane - 16));
        if (writer && eout < (long long)n_edges) {
            out[eout] = diag;
        }
    }
}

extern "C" void kernel_launch(void* const* d_in, const int* in_sizes, int n_in,
                              void* d_out, int out_size, void* d_ws, size_t ws_size,
                              hipStream_t stream) {
    const float* h   = (const float*)d_in[0];
    const int*   src = (const int*)d_in[1];
    const int*   dst = (const int*)d_in[2];
    float*       out = (float*)d_out;

    const int n_edges = in_sizes[1];
    const int tiles   = (n_edges + EDGES_PER_WAVE - 1) / EDGES_PER_WAVE;
    const int blocks  = (tiles + WAVES_PER_BLOCK - 1) / WAVES_PER_BLOCK;

    edge_dot_wmma_f32<<<blocks, 256, 0, stream>>>(h, src, dst, out, n_edges);
}